// InpatientInterventions_60644938219534
// MI455X (gfx1250) — compile-verified
//
#include <hip/hip_runtime.h>

typedef __attribute__((ext_vector_type(2))) float v2f;
typedef __attribute__((ext_vector_type(8))) float v8f;

#define TCHUNK 128  // time rows per scan chunk (8 WMMA tiles of 16)

// ---------------------------------------------------------------- zero d_out
__global__ void zero_out_kernel(float* __restrict__ out, int n) {
  int i = blockIdx.x * blockDim.x + threadIdx.x;
  int stride = gridDim.x * blockDim.x;
  for (; i < n; i += stride) out[i] = 0.0f;
}

// ------------------------------------------- per-event difference scatter
// out[t, g] accumulates +sr at t=lo and -sr at t=hi (range [lo,hi) active),
// where lo/hi are lower_bounds of start/end in the sorted t0 array.
__global__ void event_scatter_kernel(const int* __restrict__ index,
                                     const float* __restrict__ rate,
                                     const float* __restrict__ starttime,
                                     const float* __restrict__ endtime,
                                     const float* __restrict__ t0,
                                     const int* __restrict__ group_id,
                                     const float* __restrict__ weights,
                                     float* __restrict__ diff,
                                     int E, int T, int G) {
  int e = blockIdx.x * blockDim.x + threadIdx.x;
  if (e >= E) return;
  int src = index[e];
  float sr = rate[e] * weights[src];
  int g = group_id[src];
  float st = starttime[e];
  float en = endtime[e];
  // lower_bound(t0, st): first t with t0[t] >= st
  int lo = 0, hb = T;
  while (lo < hb) { int mid = (lo + hb) >> 1; if (t0[mid] < st) lo = mid + 1; else hb = mid; }
  // lower_bound(t0, en), en > st so start from lo
  int lo2 = lo, hb2 = T;
  while (lo2 < hb2) { int mid = (lo2 + hb2) >> 1; if (t0[mid] < en) lo2 = mid + 1; else hb2 = mid; }
  if (lo  < T) atomicAdd(&diff[(size_t)lo  * G + g],  sr);
  if (lo2 < T) atomicAdd(&diff[(size_t)lo2 * G + g], -sr);
}

// ---------------------------------------------- per-chunk column sums
__global__ void chunk_sum_kernel(const float* __restrict__ diff,
                                 float* __restrict__ partial, int T, int G) {
  int idx = blockIdx.x * blockDim.x + threadIdx.x;
  int total = (T / TCHUNK) * G;
  if (idx >= total) return;
  int chunk = idx / G;
  int col = idx - chunk * G;
  const float* p = diff + (size_t)chunk * TCHUNK * G + col;
  float s = 0.0f;
  for (int r = 0; r < TCHUNK; ++r) s += p[(size_t)r * G];
  partial[idx] = s;
}

// --------------------------------- exclusive scan of chunk sums per column
__global__ void scan_partial_kernel(float* __restrict__ partial, int nchunk, int G) {
  int col = blockIdx.x * blockDim.x + threadIdx.x;
  if (col >= G) return;
  float run = 0.0f;
  for (int c = 0; c < nchunk; ++c) {
    float v = partial[(size_t)c * G + col];
    partial[(size_t)c * G + col] = run;
    run += v;
  }
}

// ----------------- in-place inclusive scan of 16x16 tiles via WMMA f32
// D = L(16x16, lower-triangular ones) * X(16x16) + broadcast(carry)
// done as 4 accumulating V_WMMA_F32_16X16X4_F32 ops (K sliced 4 at a time).
__global__ void wmma_scan_kernel(float* __restrict__ data,
                                 const float* __restrict__ partial,
                                 int T, int G) {
  int lane = threadIdx.x & 31;
  int wave = threadIdx.x >> 5;
  int n    = lane & 15;      // column within tile (B/C/D layout)
  int half = lane >> 4;      // 0: lanes 0-15, 1: lanes 16-31
  int chunk = blockIdx.x;
  int cbase = (blockIdx.y * 8 + wave) * 16;
  int rbase0 = chunk * TCHUNK;

  // carry-in from exclusive scan of prior chunks, one value per column
  float carry = partial[(size_t)chunk * G + cbase + n];

  for (int tile = 0; tile < TCHUNK / 16; ++tile) {
    int rbase = rbase0 + tile * 16;
    v8f c;
#pragma unroll
    for (int i = 0; i < 8; ++i) c[i] = carry;  // 1*carry broadcast to all rows

#pragma unroll
    for (int k = 0; k < 4; ++k) {
      int kb = 4 * k + 2 * half;  // K column pair handled by this lane half
      v2f a, b;
      // A (16x4 slice of L): lanes 0-15 -> K=kb,kb+1 ; lanes 16-31 -> K=kb,kb+1 (kb+2 offset)
      a.x = (n >= kb)     ? 1.0f : 0.0f;
      a.y = (n >= kb + 1) ? 1.0f : 0.0f;
      // B (4x16 slice of X): rows rbase+4k+2*half (+0,+1), cols cbase..cbase+15
      const float* src = data + (size_t)(rbase + kb) * G + cbase + n;
      b.x = src[0];
      b.y = src[(size_t)G];
      c = __builtin_amdgcn_wmma_f32_16x16x4_f32(
          /*neg_a=*/false, a, /*neg_b=*/false, b,
          /*c_mod=*/(short)0, c, /*reuse_a=*/false, /*reuse_b=*/false);
    }

    // store inclusive-scanned tile back in place
#pragma unroll
    for (int r = 0; r < 8; ++r)
      data[(size_t)(rbase + r + 8 * half) * G + cbase + n] = c[r];

    // new carry = scanned row 15, held in c[7] of lanes 16-31
    carry = __shfl(c[7], 16 + n, 32);
  }
}

// ---------------------------------------------------------------- launcher
extern "C" void kernel_launch(void* const* d_in, const int* in_sizes, int n_in,
                              void* d_out, int out_size, void* d_ws, size_t ws_size,
                              hipStream_t stream) {
  const int*   index     = (const int*)d_in[0];
  const float* rate      = (const float*)d_in[1];
  const float* starttime = (const float*)d_in[2];
  const float* endtime   = (const float*)d_in[3];
  const float* t0        = (const float*)d_in[4];
  const int*   group_id  = (const int*)d_in[5];
  const float* weights   = (const float*)d_in[6];
  float* out = (float*)d_out;

  int E = in_sizes[0];
  int T = in_sizes[4];
  int G = out_size / T;          // 1024
  int nchunk = T / TCHUNK;       // 32

  float* partial = (float*)d_ws; // nchunk*G floats = 128 KB

  // 1. zero the difference/output buffer (rebuilt every call -> deterministic)
  zero_out_kernel<<<512, 256, 0, stream>>>(out, out_size);

  // 2. scatter +/- sr at range endpoints (2 atomics per event)
  event_scatter_kernel<<<(E + 255) / 256, 256, 0, stream>>>(
      index, rate, starttime, endtime, t0, group_id, weights, out, E, T, G);

  // 3. per-chunk column sums
  int total = nchunk * G;
  chunk_sum_kernel<<<(total + 255) / 256, 256, 0, stream>>>(out, partial, T, G);

  // 4. exclusive scan of chunk sums per column
  scan_partial_kernel<<<(G + 255) / 256, 256, 0, stream>>>(partial, nchunk, G);

  // 5. WMMA-based in-chunk inclusive scan with carry, in place
  dim3 grid(nchunk, G / 128);    // 32 x 8 blocks, 8 waves each
  wmma_scan_kernel<<<grid, 256, 0, stream>>>(out, partial, T, G);
}